// PointNet_14388140441772
// MI455X (gfx1250) — compile-verified
//
#include <hip/hip_runtime.h>
#include <hip/hip_bf16.h>

// Problem constants (from reference)
#define NV      40000
#define NP      32
#define C_OUT   64
#define KDIM    9          // 4 raw + 3 cluster + 2 center
#define KPAD    12         // pad to 3 WMMA k-steps of 4
#define VX_     0.16f
#define VY_     0.16f
#define X_OFF   (VX_ / 2.0f + 0.0f)
#define Y_OFF   (VY_ / 2.0f + -39.68f)
#define BN_EPS_ 1e-5f

typedef float v2f __attribute__((ext_vector_type(2)));
typedef float v8f __attribute__((ext_vector_type(8)));
typedef float v4f __attribute__((ext_vector_type(4)));

// ---------------------------------------------------------------------------
// Pass 1: one wave32 per voxel.
//   - build 9-ch features (masked) in LDS (K padded to 12 with zeros)
//   - x[32x64] = feat[32x12] * W^T[12x64] via 24x V_WMMA_F32_16X16X4_F32
//   - per-(v,o) max & min over the 32 points -> maxbuf/minbuf
//   - per-channel sum / sumsq -> LDS block accum -> global atomics
// ---------------------------------------------------------------------------
__global__ __launch_bounds__(256) void pfn_pass1(
    const float* __restrict__ voxels,          // (V, 32, 4)
    const int*   __restrict__ npts,            // (V,)
    const int*   __restrict__ coors,           // (V, 2)
    const float* __restrict__ W,               // (64, 9)
    float*       __restrict__ stats,           // [0..63]=sum, [64..127]=sumsq
    float*       __restrict__ maxbuf,          // (V, 64)
    float*       __restrict__ minbuf)          // (V, 64)
{
    __shared__ float feat[8][NP][KPAD];        // per-wave feature tile
    __shared__ float blk_acc[2 * C_OUT];       // block-level BN accumulators

    const int lane = threadIdx.x & 31;
    const int wave = threadIdx.x >> 5;
    const bool hi  = lane >= 16;
    const int  ln  = lane & 15;

    if (threadIdx.x < 2 * C_OUT) blk_acc[threadIdx.x] = 0.0f;
    __syncthreads();

    // ---- B fragments: Wt[k][o], k-striped within lane (mirrors A layout) ----
    // lane<16 : VGPR0=K(kt*4+0), VGPR1=K(kt*4+1), column o = nt*16+lane
    // lane>=16: VGPR0=K(kt*4+2), VGPR1=K(kt*4+3)
    v2f Bfrag[3][4];
#pragma unroll
    for (int kt = 0; kt < 3; ++kt) {
#pragma unroll
        for (int nt = 0; nt < 4; ++nt) {
            const int o  = nt * 16 + ln;
            const int k0 = kt * 4 + (hi ? 2 : 0);
            const float b0 = (k0     < KDIM) ? W[o * KDIM + k0    ] : 0.0f;
            const float b1 = (k0 + 1 < KDIM) ? W[o * KDIM + k0 + 1] : 0.0f;
            v2f b; b.x = b0; b.y = b1;
            Bfrag[kt][nt] = b;
        }
    }

    const int v = blockIdx.x * 8 + wave;       // grid = V/8 exactly

    // ---- features: lane == point index p ----
    const v4f pt = *(const v4f*)(voxels + (((size_t)v * NP) + lane) * 4);
    float x = pt.x, y = pt.y, z = pt.z, wf = pt.w;

    // xyz mean: sum over ALL 32 points / npts (matches reference)
    float sx = x, sy = y, sz = z;
#pragma unroll
    for (int m = 16; m >= 1; m >>= 1) {
        sx += __shfl_xor(sx, m, 32);
        sy += __shfl_xor(sy, m, 32);
        sz += __shfl_xor(sz, m, 32);
    }
    const int   n    = npts[v];
    const float invn = 1.0f / (float)n;
    const float mx = sx * invn, my = sy * invn, mz = sz * invn;
    const float cx = (float)coors[v * 2 + 0] * VX_ + X_OFF;
    const float cy = (float)coors[v * 2 + 1] * VY_ + Y_OFF;
    const float msk = (lane < n) ? 1.0f : 0.0f;

    float f[KPAD];
    f[0] = x;      f[1] = y;      f[2] = z;      f[3] = wf;
    f[4] = x - mx; f[5] = y - my; f[6] = z - mz;
    f[7] = x - cx; f[8] = y - cy;
    f[9] = 0.0f;   f[10] = 0.0f;  f[11] = 0.0f;
#pragma unroll
    for (int k = 0; k < KDIM; ++k) feat[wave][lane][k] = f[k] * msk;
#pragma unroll
    for (int k = KDIM; k < KPAD; ++k) feat[wave][lane][k] = 0.0f;
    // same-wave LDS RAW: backend inserts s_wait_dscnt

    // ---- GEMM: x[32x64] = feat * W^T, 2x4x3 = 24 WMMAs ----
    v8f acc[2][4] = {};
#pragma unroll
    for (int mt = 0; mt < 2; ++mt) {
        const int row = mt * 16 + ln;
        v2f Afrag[3];
#pragma unroll
        for (int kt = 0; kt < 3; ++kt) {
            const int k0 = kt * 4 + (hi ? 2 : 0);
            v2f a; a.x = feat[wave][row][k0]; a.y = feat[wave][row][k0 + 1];
            Afrag[kt] = a;
        }
#pragma unroll
        for (int nt = 0; nt < 4; ++nt) {
#pragma unroll
            for (int kt = 0; kt < 3; ++kt) {
                acc[mt][nt] = __builtin_amdgcn_wmma_f32_16x16x4_f32(
                    false, Afrag[kt], false, Bfrag[kt][nt],
                    (short)0, acc[mt][nt], false, false);
            }
        }
    }

    // ---- reductions over the 32 points per output channel ----
    // D layout: tile(mt,nt), VGPR j -> row mt*16 + j + (hi?8:0), col nt*16+ln.
    // A lane covers 16 of 32 rows for its column; lane^16 has the other 16.
#pragma unroll
    for (int nt = 0; nt < 4; ++nt) {
        float vmax = -3.4e38f, vmin = 3.4e38f, s = 0.0f, sq = 0.0f;
#pragma unroll
        for (int mt = 0; mt < 2; ++mt) {
#pragma unroll
            for (int j = 0; j < 8; ++j) {
                const float t = acc[mt][nt][j];
                vmax = fmaxf(vmax, t);
                vmin = fminf(vmin, t);
                s  += t;
                sq += t * t;
            }
        }
        vmax = fmaxf(vmax, __shfl_xor(vmax, 16, 32));
        vmin = fminf(vmin, __shfl_xor(vmin, 16, 32));
        s  += __shfl_xor(s,  16, 32);
        sq += __shfl_xor(sq, 16, 32);

        if (lane < 16) {
            const int o = nt * 16 + ln;
            maxbuf[(size_t)v * C_OUT + o] = vmax;
            minbuf[(size_t)v * C_OUT + o] = vmin;
            atomicAdd(&blk_acc[o], s);
            atomicAdd(&blk_acc[C_OUT + o], sq);
        }
    }

    __syncthreads();
    if (threadIdx.x < 2 * C_OUT)
        atomicAdd(&stats[threadIdx.x], blk_acc[threadIdx.x]);
}

// ---------------------------------------------------------------------------
// Pass 2: BN affine + ReLU + maxpool, using monotonicity:
//   max_p relu(a*x+b) = relu(a * (a>=0 ? max_p x : min_p x) + b)
// ---------------------------------------------------------------------------
__global__ __launch_bounds__(256) void pfn_pass2(
    const float* __restrict__ stats,
    const float* __restrict__ maxbuf,
    const float* __restrict__ minbuf,
    const float* __restrict__ gamma,
    const float* __restrict__ beta,
    float*       __restrict__ out)
{
    const int idx = blockIdx.x * blockDim.x + threadIdx.x;
    if (idx >= NV * C_OUT) return;
    const int o = idx & (C_OUT - 1);

    const float inv_cnt = 1.0f / ((float)NV * (float)NP);
    const float mean = stats[o] * inv_cnt;
    const float var  = stats[C_OUT + o] * inv_cnt - mean * mean;
    const float a    = gamma[o] * rsqrtf(var + BN_EPS_);
    const float b    = beta[o] - mean * a;

    const float xm = (a >= 0.0f) ? maxbuf[idx] : minbuf[idx];
    out[idx] = fmaxf(0.0f, a * xm + b);
}

extern "C" void kernel_launch(void* const* d_in, const int* in_sizes, int n_in,
                              void* d_out, int out_size, void* d_ws, size_t ws_size,
                              hipStream_t stream) {
    const float* voxels = (const float*)d_in[0];
    const int*   npts   = (const int*)  d_in[1];
    const int*   coors  = (const int*)  d_in[2];
    const float* W      = (const float*)d_in[3];
    const float* gamma  = (const float*)d_in[4];
    const float* beta   = (const float*)d_in[5];
    float* out = (float*)d_out;

    float* ws     = (float*)d_ws;
    float* stats  = ws;                                // 128 floats
    float* maxbuf = ws + 128;                          // V*64 floats
    float* minbuf = ws + 128 + (size_t)NV * C_OUT;     // V*64 floats

    hipMemsetAsync(stats, 0, 2 * C_OUT * sizeof(float), stream);
    pfn_pass1<<<NV / 8, 256, 0, stream>>>(voxels, npts, coors, W,
                                          stats, maxbuf, minbuf);
    pfn_pass2<<<(NV * C_OUT + 255) / 256, 256, 0, stream>>>(
        stats, maxbuf, minbuf, gamma, beta, out);
}